// BidirectionalMambaBlock_17832704213069
// MI455X (gfx1250) — compile-verified
//
#include <hip/hip_runtime.h>
#include <math.h>

#define BATCH  4
#define SEQ    2048
#define DMODEL 256
#define DFF    1024
#define DSTATE 16
#define DCONV  4
#define DINNER 512
#define DTRANK 16
#define NROWS  (BATCH * SEQ)   /* 8192 */
#define EPSV   1e-5f

typedef float v2f __attribute__((ext_vector_type(2)));
typedef float v8f __attribute__((ext_vector_type(8)));

// ---------------------------------------------------------------------------
// Register-blocked fp32 WMMA GEMM:  C[M,N] = epilogue( A[M,K](lda) @ W[N,K]^T )
// Each wave computes a (16*MT) x (16*NT) macro-tile via V_WMMA_F32_16X16X4_F32,
// reusing each A fragment NT times and each B fragment MT times; the MT*NT
// WMMAs per K-step hit independent accumulators (no D->C serialization).
// MODE: 0 = none, 1 = softplus(x + bias), 2 = relu(x + bias), 3 = x + bias
// Requires M % (64*MT) == 0, N % (16*NT) == 0, K % 4 == 0 (true at all call
// sites). No divergence anywhere (EXEC all ones for WMMA).
// ---------------------------------------------------------------------------
template <int MT, int NT, int MODE>
__global__ __launch_bounds__(128) void gemm_wmma_f32(
    const float* __restrict__ A, const float* __restrict__ W,
    float* __restrict__ C, const float* __restrict__ bias,
    int M, int N, int K, int lda)
{
  const int lane = threadIdx.x;        // 0..31
  const int half = lane >> 4;          // K sub-pair select (K+0/1 vs K+2/3)
  const int l16  = lane & 15;
  const int n0   = blockIdx.x * (16 * NT);
  const int m0   = (blockIdx.y * 4 + threadIdx.y) * (16 * MT);

  const float* arow[MT];
  const float* wrow[NT];
#pragma unroll
  for (int i = 0; i < MT; ++i)
    arow[i] = A + (size_t)(m0 + 16 * i + l16) * lda + half * 2;
#pragma unroll
  for (int j = 0; j < NT; ++j)
    wrow[j] = W + (size_t)(n0 + 16 * j + l16) * K + half * 2;

  v8f acc[MT][NT];
#pragma unroll
  for (int i = 0; i < MT; ++i)
#pragma unroll
    for (int j = 0; j < NT; ++j) acc[i][j] = (v8f){};

  for (int k = 0; k < K; k += 4) {
    v2f av[MT], bv[NT];
#pragma unroll
    for (int i = 0; i < MT; ++i) {
      float2 t = *(const float2*)(arow[i] + k);
      av[i].x = t.x; av[i].y = t.y;
    }
#pragma unroll
    for (int j = 0; j < NT; ++j) {
      float2 t = *(const float2*)(wrow[j] + k);
      bv[j].x = t.x; bv[j].y = t.y;
    }
#pragma unroll
    for (int i = 0; i < MT; ++i)
#pragma unroll
      for (int j = 0; j < NT; ++j)
        acc[i][j] = __builtin_amdgcn_wmma_f32_16x16x4_f32(
            false, av[i], false, bv[j], (short)0, acc[i][j], false, false);
  }

#pragma unroll
  for (int j = 0; j < NT; ++j) {
    const int col = n0 + 16 * j + l16;
    float bz = (MODE != 0) ? bias[col] : 0.0f;
#pragma unroll
    for (int i = 0; i < MT; ++i) {
#pragma unroll
      for (int v = 0; v < 8; ++v) {
        int row = m0 + 16 * i + v + half * 8;  // D layout: VGPR v -> M=v / v+8
        float val = acc[i][j][v];
        if (MODE != 0) val += bz;
        if (MODE == 1) val = (val > 20.0f) ? val : logf(1.0f + __expf(val));
        else if (MODE == 2) val = fmaxf(val, 0.0f);
        C[(size_t)row * N + col] = val;
      }
    }
  }
}

// ---------------------------------------------------------------------------
// Causal depthwise conv (D_CONV=4) + bias + SiLU. dir=1 mirrors tap offsets so
// everything stays in physical time order (equivalent to conv on the flipped
// sequence, results stored at physical positions).
// ---------------------------------------------------------------------------
__global__ __launch_bounds__(256) void conv_silu(
    const float* __restrict__ xz, const float* __restrict__ cw,
    const float* __restrict__ cb, float* __restrict__ xi, int dir)
{
  int idx = blockIdx.x * blockDim.x + threadIdx.x;
  if (idx >= NROWS * DINNER) return;
  int c   = idx % DINNER;
  int row = idx / DINNER;      // b*SEQ + t
  int t   = row % SEQ;
  int b   = row / SEQ;
  float s = cb[c];
#pragma unroll
  for (int j = 0; j < DCONV; ++j) {
    int tp = dir ? (t + (DCONV - 1) - j) : (t - (DCONV - 1) + j);
    if (tp >= 0 && tp < SEQ)
      s += cw[c * DCONV + j] * xz[((size_t)(b * SEQ + tp)) * (2 * DINNER) + c];
  }
  float sig = 1.0f / (1.0f + __expf(-s));
  xi[(size_t)row * DINNER + c] = s * sig;
}

// ---------------------------------------------------------------------------
// Selective-scan: one lane per (batch, inner-channel); h[16] in registers,
// serial over logical time (dir-aware physical index). Fused D-skip and
// SiLU(z) gating on output; output stored at physical positions (unflipped).
// ---------------------------------------------------------------------------
__global__ __launch_bounds__(256) void ssm_scan(
    const float* __restrict__ xz, const float* __restrict__ xi,
    const float* __restrict__ dbc, const float* __restrict__ delta,
    const float* __restrict__ Alog, const float* __restrict__ Dsk,
    float* __restrict__ y, int dir)
{
  int tid = blockIdx.x * blockDim.x + threadIdx.x;
  if (tid >= BATCH * DINNER) return;
  int d = tid % DINNER;
  int b = tid / DINNER;

  float Arow[DSTATE], h[DSTATE];
#pragma unroll
  for (int n = 0; n < DSTATE; ++n) {
    Arow[n] = -__expf(Alog[d * DSTATE + n]);   // A = -exp(Alog)
    h[n] = 0.0f;
  }
  float Dv = Dsk[d];

  for (int t = 0; t < SEQ; ++t) {
    int p = dir ? (SEQ - 1 - t) : t;
    size_t row = (size_t)b * SEQ + p;
    float u  = xi[row * DINNER + d];
    float dl = delta[row * DINNER + d];
    const float* bc = dbc + row * (DTRANK + 2 * DSTATE) + DTRANK;
    float du  = dl * u;
    float acc = 0.0f;
#pragma unroll
    for (int n = 0; n < DSTATE; ++n) {
      float dA = __expf(dl * Arow[n]);
      h[n] = dA * h[n] + du * bc[n];           // B at bc[0..15]
      acc += h[n] * bc[DSTATE + n];            // C at bc[16..31]
    }
    float zv = xz[row * (2 * DINNER) + DINNER + d];
    float g  = zv / (1.0f + __expf(-zv));      // silu(z)
    y[row * DINNER + d] = (acc + u * Dv) * g;
  }
}

// ---------------------------------------------------------------------------
// Fused: BN(dir outputs) -> AddNorm LayerNorms -> sum. One block per row.
// ---------------------------------------------------------------------------
__device__ __forceinline__ float block_sum(float v, float* red) {
  int t = threadIdx.x;
  red[t] = v; __syncthreads();
  for (int s = 128; s > 0; s >>= 1) {
    if (t < s) red[t] += red[t + s];
    __syncthreads();
  }
  float r = red[0]; __syncthreads();
  return r;
}

__global__ __launch_bounds__(256) void combine_ln(
    const float* __restrict__ x, const float* __restrict__ f0,
    const float* __restrict__ f1,
    const float* __restrict__ bng, const float* __restrict__ bnb,
    const float* __restrict__ bnm, const float* __restrict__ bnv,
    const float* __restrict__ lng, const float* __restrict__ lnb,
    float* __restrict__ out)
{
  __shared__ float red[256];
  int row = blockIdx.x;
  int c   = threadIdx.x;
  size_t i = (size_t)row * DMODEL + c;
  float xv = x[i];
  float a = xv + (f0[i] - bnm[c]) * rsqrtf(bnv[c] + EPSV) * bng[c] + bnb[c];
  float b = xv + (f1[i] - bnm[DMODEL + c]) * rsqrtf(bnv[DMODEL + c] + EPSV)
                 * bng[DMODEL + c] + bnb[DMODEL + c];
  float sa  = block_sum(a, red);
  float sa2 = block_sum(a * a, red);
  float sb  = block_sum(b, red);
  float sb2 = block_sum(b * b, red);
  float ma = sa / DMODEL, va = sa2 / DMODEL - ma * ma;
  float mb = sb / DMODEL, vb = sb2 / DMODEL - mb * mb;
  float la = (a - ma) * rsqrtf(va + EPSV) * lng[c] + lnb[c];
  float lb = (b - mb) * rsqrtf(vb + EPSV) * lng[DMODEL + c] + lnb[DMODEL + c];
  out[i] = la + lb;
}

// out = BN3(ff_out) + x
__global__ __launch_bounds__(256) void final_bn_add(
    const float* __restrict__ ff, const float* __restrict__ x,
    const float* __restrict__ g, const float* __restrict__ bta,
    const float* __restrict__ m, const float* __restrict__ v,
    float* __restrict__ out)
{
  int idx = blockIdx.x * blockDim.x + threadIdx.x;
  if (idx >= NROWS * DMODEL) return;
  int c = idx % DMODEL;
  out[idx] = (ff[idx] - m[c]) * rsqrtf(v[c] + EPSV) * g[c] + bta[c] + x[idx];
}

// ---------------------------------------------------------------------------
extern "C" void kernel_launch(void* const* d_in, const int* in_sizes, int n_in,
                              void* d_out, int out_size, void* d_ws, size_t ws_size,
                              hipStream_t stream) {
  const float* x     = (const float*)d_in[0];
  const float* in_w  = (const float*)d_in[1];   // (2, 1024, 256)
  const float* cw    = (const float*)d_in[2];   // (2, 512, 4)
  const float* cb    = (const float*)d_in[3];   // (2, 512)
  const float* xpw   = (const float*)d_in[4];   // (2, 48, 512)
  const float* dtw   = (const float*)d_in[5];   // (2, 512, 16)
  const float* dtb   = (const float*)d_in[6];   // (2, 512)
  const float* Alog  = (const float*)d_in[7];   // (2, 512, 16)
  const float* Dsk   = (const float*)d_in[8];   // (2, 512)
  const float* outw  = (const float*)d_in[9];   // (2, 256, 512)
  const float* bng   = (const float*)d_in[10];  // (3, 256)
  const float* bnb   = (const float*)d_in[11];
  const float* bnm   = (const float*)d_in[12];
  const float* bnv   = (const float*)d_in[13];
  const float* lng   = (const float*)d_in[14];  // (3, 256)
  const float* lnb   = (const float*)d_in[15];
  const float* ffw1  = (const float*)d_in[16];  // (1024, 256)
  const float* ffb1  = (const float*)d_in[17];
  const float* ffw2  = (const float*)d_in[18];  // (256, 1024)
  const float* ffb2  = (const float*)d_in[19];
  float* out = (float*)d_out;

  char* ws = (char*)d_ws;
  size_t off = 0;
  auto alloc = [&](size_t bytes) -> float* {
    float* p = (float*)(ws + off);
    off += (bytes + 255) & ~(size_t)255;
    return p;
  };
  float* xzb  = alloc((size_t)NROWS * 1024 * 4);  // in_proj out (xi | z)
  float* xib  = alloc((size_t)NROWS * 512 * 4);   // silu(conv(xi))
  float* dbcb = alloc((size_t)NROWS * 48 * 4);    // dt | B | C
  float* dlt  = alloc((size_t)NROWS * 512 * 4);   // softplus delta
  float* yb   = alloc((size_t)NROWS * 512 * 4);   // gated scan output
  float* o0   = alloc((size_t)NROWS * 256 * 4);   // fwd mamba out
  float* o1   = alloc((size_t)NROWS * 256 * 4);   // bwd mamba out
  float* sum  = alloc((size_t)NROWS * 256 * 4);   // ln(f)+ln(bwd)
  float* ffh  = alloc((size_t)NROWS * 1024 * 4);  // FFN hidden
  float* ffo  = alloc((size_t)NROWS * 256 * 4);   // FFN out

  dim3 gblk(32, 4);                 // 4 waves per block
  const int MB = NROWS / (64 * 2);  // grid.y for MT=2 macro-tiles (64)

  for (int dir = 0; dir < 2; ++dir) {
    // in_proj: (8192,256) @ (1024,256)^T -> xz   [32x32 tiles]
    gemm_wmma_f32<2, 2, 0><<<dim3(1024 / 32, MB), gblk, 0, stream>>>(
        x, in_w + (size_t)dir * 1024 * 256, xzb, nullptr, NROWS, 1024, 256, 256);
    // depthwise causal conv + silu
    conv_silu<<<(NROWS * 512 + 255) / 256, 256, 0, stream>>>(
        xzb, cw + dir * 512 * 4, cb + dir * 512, xib, dir);
    // x_proj: (8192,512) @ (48,512)^T -> dt|B|C   [32x48 tiles, A reused 3x]
    gemm_wmma_f32<2, 3, 0><<<dim3(1, MB), gblk, 0, stream>>>(
        xib, xpw + (size_t)dir * 48 * 512, dbcb, nullptr, NROWS, 48, 512, 512);
    // dt_proj + softplus: (8192,16 of 48) @ (512,16)^T -> delta
    gemm_wmma_f32<2, 2, 1><<<dim3(512 / 32, MB), gblk, 0, stream>>>(
        dbcb, dtw + (size_t)dir * 512 * 16, dlt, dtb + dir * 512,
        NROWS, 512, 16, 48);
    // selective scan + D-skip + silu(z) gate
    ssm_scan<<<(BATCH * DINNER + 255) / 256, 256, 0, stream>>>(
        xzb, xib, dbcb, dlt, Alog + (size_t)dir * 512 * 16, Dsk + dir * 512,
        yb, dir);
    // out_proj: (8192,512) @ (256,512)^T
    gemm_wmma_f32<2, 2, 0><<<dim3(256 / 32, MB), gblk, 0, stream>>>(
        yb, outw + (size_t)dir * 256 * 512, dir ? o1 : o0, nullptr,
        NROWS, 256, 512, 512);
  }

  // BN0/BN1 + AddNorm LNs + sum
  combine_ln<<<NROWS, 256, 0, stream>>>(x, o0, o1, bng, bnb, bnm, bnv,
                                        lng, lnb, sum);
  // FFN: relu(sum @ w1^T + b1) @ w2^T + b2
  gemm_wmma_f32<2, 2, 2><<<dim3(1024 / 32, MB), gblk, 0, stream>>>(
      sum, ffw1, ffh, ffb1, NROWS, 1024, 256, 256);
  gemm_wmma_f32<2, 2, 3><<<dim3(256 / 32, MB), gblk, 0, stream>>>(
      ffh, ffw2, ffo, ffb2, NROWS, 256, 1024, 1024);
  // BN2 + residual
  final_bn_add<<<(NROWS * 256 + 255) / 256, 256, 0, stream>>>(
      ffo, x, bng + 2 * 256, bnb + 2 * 256, bnm + 2 * 256, bnv + 2 * 256, out);

  (void)in_sizes; (void)n_in; (void)out_size; (void)ws_size;
}